// NMSquaredGaussianMixture_61323543053025
// MI455X (gfx1250) — compile-verified
//
#include <hip/hip_runtime.h>
#include <hip/hip_bf16.h>
#include <math.h>

typedef __attribute__((ext_vector_type(2))) float v2f;
typedef __attribute__((ext_vector_type(8))) float v8f;

#define TWO_PI_D 39.478417604357434f   /* (2*pi)^2 for d=2 */
#define NEG_HALF_LOG2E (-0.7213475204444817f)  /* -0.5 * log2(e) */

// ---------------------------------------------------------------------------
// Kernel 1: per-cluster preprocessing (K=16, d=2) + pairwise normalizer z.
// coef[j][k] are pre-scaled so the WMMA output is directly the base-2
// exponent:  m'_k(x) = -0.5*log2(e)*maha_k(x) + log2|w_k/norm_k|.
// sgn[k] carries the sign of w_k.
// ---------------------------------------------------------------------------
__global__ void gmm_setup(const float* __restrict__ means,
                          const float* __restrict__ chols,
                          const float* __restrict__ weights,
                          float* __restrict__ coef,   // [8][16] row-major
                          float* __restrict__ sgn,    // [16]
                          double* __restrict__ zptr,
                          double* __restrict__ acc)
{
    if (threadIdx.x != 0 || blockIdx.x != 0) return;
    float S00[16], S01[16], S11[16];
    for (int k = 0; k < 16; ++k) {
        // L = tril(chols_k):  [L00 0; L10 L11]
        float L00 = chols[k*4 + 0];
        float L10 = chols[k*4 + 2];
        float L11 = chols[k*4 + 3];
        // S = L L^T
        float s00 = L00*L00;
        float s01 = L00*L10;
        float s11 = L10*L10 + L11*L11;
        S00[k] = s00; S01[k] = s01; S11[k] = s11;
        float det = s00*s11 - s01*s01;
        // Q = S^-1
        float q00 =  s11/det, q01 = -s01/det, q11 = s00/det;
        float mu0 = means[k*2+0], mu1 = means[k*2+1];
        float p0 = q00*mu0 + q01*mu1;   // (Q mu)_0
        float p1 = q01*mu0 + q11*mu1;   // (Q mu)_1
        float cc = mu0*p0 + mu1*p1;     // mu^T Q mu
        const float s = NEG_HALF_LOG2E;
        float wl = weights[k] / sqrtf(TWO_PI_D * det);
        // exponent(x) = feat(x) . coef[:,k],  feat = [x0^2,x0x1,x1^2,x0,x1,1]
        coef[0*16+k] = s * q00;
        coef[1*16+k] = s * 2.0f*q01;
        coef[2*16+k] = s * q11;
        coef[3*16+k] = s * -2.0f*p0;
        coef[4*16+k] = s * -2.0f*p1;
        coef[5*16+k] = s * cc + log2f(fabsf(wl));
        coef[6*16+k] = 0.0f;
        coef[7*16+k] = 0.0f;
        sgn[k] = (wl < 0.0f) ? -1.0f : 1.0f;
    }
    // z = sum_ij w_i w_j N(mu_i - mu_j; 0, S_i + S_j)
    double z = 0.0;
    for (int i = 0; i < 16; ++i) {
        for (int j = 0; j < 16; ++j) {
            float a00 = S00[i]+S00[j], a01 = S01[i]+S01[j], a11 = S11[i]+S11[j];
            float det = a00*a11 - a01*a01;
            float d0 = means[i*2+0]-means[j*2+0];
            float d1 = means[i*2+1]-means[j*2+1];
            float m2 = (a11*d0*d0 - 2.0f*a01*d0*d1 + a00*d1*d1) / det;
            double Z = exp(-0.5*(double)m2) / sqrt((double)(TWO_PI_D*det));
            z += (double)weights[i] * (double)weights[j] * Z;
        }
    }
    *zptr = z;
    *acc  = 0.0;
}

// ---------------------------------------------------------------------------
// Kernel 2: main pass. One wave32 handles a tile of 16 samples.
// GEMM orientation: M = cluster (A operand, constant), N = sample (B operand).
// D[M=cluster, N=sample]: each lane holds 8 clusters of ONE sample
// (lanes 0-15 -> clusters 0-7, lanes 16-31 -> clusters 8-15), so the
// cluster reduction is an in-lane tree + a single xor-16 shuffle.
// Trip count is made wave-uniform via readfirstlane -> pure scalar loop ctrl.
// ---------------------------------------------------------------------------
__global__ void __launch_bounds__(256)
gmm_main(const float* __restrict__ X,
         const float* __restrict__ coef,
         const float* __restrict__ sgn,
         double* __restrict__ acc,
         int ntiles)
{
    const int lane = threadIdx.x & 31;
    const int n16  = lane & 15;
    const bool hi  = lane >= 16;
    const int wavesPerBlock = blockDim.x >> 5;
    const int gwave  = blockIdx.x * wavesPerBlock + (threadIdx.x >> 5);
    const int nwaves = gridDim.x * wavesPerBlock;

    // A operand (16x4, M=cluster x K=feature), constant across tiles.
    // Lanes 0-15 hold K=0,1; lanes 16-31 hold K=2,3 (cluster = lane&15).
    const float c0 = coef[0*16+n16], c1 = coef[1*16+n16], c2 = coef[2*16+n16];
    const float c3 = coef[3*16+n16], c4 = coef[4*16+n16], c5 = coef[5*16+n16];
    v2f a0, a1;
    a0[0] = hi ? c2 : c0;
    a0[1] = hi ? c3 : c1;
    a1[0] = hi ? 0.0f : c4;
    a1[1] = hi ? 0.0f : c5;

    // Per-lane signs for the 8 clusters this lane's accumulators hold.
    float sgn8[8];
    #pragma unroll
    for (int v = 0; v < 8; ++v)
        sgn8[v] = hi ? sgn[v + 8] : sgn[v];

    // Wave-uniform trip count in an SGPR -> scalar loop control, no EXEC mask.
    int myTiles = (gwave < ntiles) ? ((ntiles - 1 - gwave) / nwaves + 1) : 0;
    myTiles = __builtin_amdgcn_readfirstlane(myTiles);

    // Running pointer; constant stride per grid-stride step.
    const float* xp = X + ((long long)gwave * 16 + n16) * 2;
    const long long stride = (long long)nwaves * 32;   // floats per step

    double accd = 0.0;
    for (int i = 0; i < myTiles; ++i) {
        const float x0 = xp[0];
        const float x1 = xp[1];
        // speculative prefetch of this wave's next tile into near caches
        __builtin_prefetch(xp + stride, 0, 3);
        // B operand (4x16, K=feature x N=sample):
        // lanes 0-15 hold K=0,1; lanes 16-31 hold K=2,3 (sample = lane&15).
        v2f b0, b1;
        b0[0] = hi ? x1*x1 : x0*x0;   // K2 : K0
        b0[1] = hi ? x0    : x0*x1;   // K3 : K1
        b1[0] = hi ? 0.0f  : x1;      // pad : K4
        b1[1] = hi ? 0.0f  : 1.0f;    // pad : K5
        v8f m = {};
        m = __builtin_amdgcn_wmma_f32_16x16x4_f32(false, a0, false, b0,
                                                  (short)0, m, false, false);
        m = __builtin_amdgcn_wmma_f32_16x16x4_f32(false, a1, false, b1,
                                                  (short)0, m, false, false);
        // m[v] is already the base-2 exponent incl. log2|w/norm|.
        float t[8];
        #pragma unroll
        for (int v = 0; v < 8; ++v)
            t[v] = sgn8[v] * __builtin_amdgcn_exp2f(m[v]);  // bare v_exp_f32
        // pairwise tree over this lane's 8 clusters
        const float p = ((t[0] + t[1]) + (t[2] + t[3]))
                      + ((t[4] + t[5]) + (t[6] + t[7]));
        // combine the two 8-cluster halves for this sample: one shuffle
        const float g = p + __shfl_xor(p, 16);
        accd += (double)g * (double)g;          // counted twice per sample
        xp += stride;
    }
    // wave-level reduction, then one atomic per wave
    #pragma unroll
    for (int off = 16; off >= 1; off >>= 1)
        accd += __shfl_down(accd, off);
    if (lane == 0) atomicAdd(acc, accd);
}

// ---------------------------------------------------------------------------
// Kernel 3: finalize  out = -log( (sum g^2 / 2) / z ) / N
// ---------------------------------------------------------------------------
__global__ void gmm_finalize(const double* __restrict__ acc,
                             const double* __restrict__ zptr,
                             float* __restrict__ out,
                             long long n)
{
    if (threadIdx.x == 0 && blockIdx.x == 0) {
        double sumg2  = *acc * 0.5;       // each sample counted twice per wave
        double pdfsum = sumg2 / *zptr;
        out[0] = (float)(-log(pdfsum) / (double)n);
    }
}

extern "C" void kernel_launch(void* const* d_in, const int* in_sizes, int n_in,
                              void* d_out, int out_size, void* d_ws, size_t ws_size,
                              hipStream_t stream)
{
    const float* X       = (const float*)d_in[0];   // [N,2]
    const float* means   = (const float*)d_in[1];   // [16,2]
    const float* chols   = (const float*)d_in[2];   // [16,2,2]
    const float* weights = (const float*)d_in[3];   // [16]
    (void)n_in; (void)ws_size; (void)out_size;

    const long long N = (long long)in_sizes[0] / 2;
    const int ntiles  = (int)(N / 16);              // N is a multiple of 16

    // workspace layout
    float*  coef = (float*)d_ws;                          // 128 floats (512 B)
    float*  sgn  = coef + 128;                            // 16 floats
    double* zp   = (double*)((char*)d_ws + 576);
    double* acc  = (double*)((char*)d_ws + 584);

    gmm_setup<<<1, 1, 0, stream>>>(means, chols, weights, coef, sgn, zp, acc);

    const int block = 256;                                // 8 wave32s
    const int grid  = 2048;                               // grid-stride over tiles
    gmm_main<<<grid, block, 0, stream>>>(X, coef, sgn, acc, ntiles);

    gmm_finalize<<<1, 1, 0, stream>>>(acc, zp, (float*)d_out, N);
}